// VectorQuantizer_79293686219022
// MI455X (gfx1250) — compile-verified
//
#include <hip/hip_runtime.h>
#include <hip/hip_bf16.h>

typedef __attribute__((ext_vector_type(2))) float v2f;
typedef __attribute__((ext_vector_type(8))) float v8f;

#define K_DIM      256        // embedding dim
#define N_EMB      1024       // codebook size
#define N_ROWS     32768      // 32 * 32 * 32 flattened spatial rows
#define N_ELEMS    8388608    // 32 * 256 * 32 * 32
#define ROWS_PER_BLK 32       // two 16-row WMMA tiles per block
#define LDS_STRIDE 260        // 256 + 4 pad -> conflict-free A fragments

// ---------------------------------------------------------------------------
// Kernel 0: per-code squared norms ||emb_k||^2, and zero the loss accumulator.
// ---------------------------------------------------------------------------
__global__ __launch_bounds__(256) void vq_embsq_kernel(
    const float* __restrict__ emb, float* __restrict__ embsq,
    float* __restrict__ loss_slot) {
  int n = blockIdx.x * blockDim.x + threadIdx.x;
  if (n == 0) *loss_slot = 0.0f;
  if (n < N_EMB) {
    const float4* row = (const float4*)(emb + (size_t)n * K_DIM);
    float s = 0.0f;
#pragma unroll 8
    for (int i = 0; i < K_DIM / 4; ++i) {
      float4 v = row[i];
      s += v.x * v.x + v.y * v.y + v.z * v.z + v.w * v.w;
    }
    embsq[n] = s;
  }
}

// ---------------------------------------------------------------------------
// Kernel 1: nearest-code search via f32 WMMA distance GEMM + argmin.
// Block = 128 threads (4 waves). Block handles 32 z rows vs all 1024 codes.
// Each B fragment (codebook, L2-resident) feeds 2 WMMAs (rows 0-15 / 16-31),
// halving L2 codebook traffic versus a 16-row tile.
// dist(m,n) = ||e_n||^2 - 2 * dot(z_m, e_n)   (||z_m||^2 constant per row)
// ---------------------------------------------------------------------------
__global__ __launch_bounds__(128) void vq_argmin_kernel(
    const float* __restrict__ z, const float* __restrict__ emb,
    const float* __restrict__ embsq, int* __restrict__ idx_out,
    float* __restrict__ idxf_out) {
  __shared__ float lds_z[ROWS_PER_BLK * LDS_STRIDE];
  __shared__ float red_val[4][ROWS_PER_BLK];
  __shared__ int   red_idx[4][ROWS_PER_BLK];

  const int tile = blockIdx.x;   // 0..1023, 32 rows each
  const int tid  = threadIdx.x;  // 0..127
  const int wave = tid >> 5;
  const int lane = tid & 31;
  const int half = lane >> 4;    // wave32: K split across lane halves
  const int lm   = lane & 15;

  // ---- stage A tile (32 rows x 256 features) into LDS -------------------
  // z layout [B,C,H,W]: z_flat[n][c] = z[b*262144 + c*1024 + (n & 1023)],
  // and within one 32-row tile b and the low-bit base are constant, so
  // consecutive m are consecutive addresses (coalesced).
  const int  n0   = tile * ROWS_PER_BLK;
  const long base = (long)(n0 >> 10) * 262144 + (long)(n0 & 1023);
#pragma unroll
  for (int it = 0; it < 64; ++it) {
    int flat = it * 128 + tid;   // 0..8191
    int m = flat & 31;
    int c = flat >> 5;
    lds_z[m * LDS_STRIDE + c] = z[base + (long)c * 1024 + m];
  }
  __syncthreads();

  // ---- per-lane running argmin over this wave's 16 column tiles ----------
  float bestv[2][8];
  int   besti[2][8];
#pragma unroll
  for (int r = 0; r < 2; ++r)
#pragma unroll
    for (int v = 0; v < 8; ++v) { bestv[r][v] = 3.0e38f; besti[r][v] = 0; }

  // A fragments: lane = M row; row-tile 0 = rows 0-15, row-tile 1 = 16-31.
  const float* ap0 = lds_z + lm * LDS_STRIDE + 2 * half;
  const float* ap1 = lds_z + (16 + lm) * LDS_STRIDE + 2 * half;

  for (int t = 0; t < 4; ++t) {
    const int ntb = wave * 16 + t * 4;  // 4 column tiles per group
    // B fragment base: lane = N (code index), K halves split across lanes.
    const float* b0 = emb + (size_t)((ntb + 0) * 16 + lm) * K_DIM + 2 * half;
    const float* b1 = emb + (size_t)((ntb + 1) * 16 + lm) * K_DIM + 2 * half;
    const float* b2 = emb + (size_t)((ntb + 2) * 16 + lm) * K_DIM + 2 * half;
    const float* b3 = emb + (size_t)((ntb + 3) * 16 + lm) * K_DIM + 2 * half;
    if (t < 3) {  // speculative prefetch of next group's B rows (L2-resident)
      __builtin_prefetch(b0 + 4 * 16 * K_DIM, 0, 0);
      __builtin_prefetch(b2 + 4 * 16 * K_DIM, 0, 0);
    }

    v8f acc[2][4] = {};
#pragma unroll 4
    for (int kk = 0; kk < 64; ++kk) {  // reduce over D=256, 4 per WMMA
      v2f a0  = *(const v2f*)(ap0 + kk * 4);
      v2f a1  = *(const v2f*)(ap1 + kk * 4);
      v2f vb0 = *(const v2f*)(b0 + kk * 4);
      v2f vb1 = *(const v2f*)(b1 + kk * 4);
      v2f vb2 = *(const v2f*)(b2 + kk * 4);
      v2f vb3 = *(const v2f*)(b3 + kk * 4);
      acc[0][0] = __builtin_amdgcn_wmma_f32_16x16x4_f32(
          false, a0, false, vb0, (short)0, acc[0][0], false, false);
      acc[1][0] = __builtin_amdgcn_wmma_f32_16x16x4_f32(
          false, a1, false, vb0, (short)0, acc[1][0], false, false);
      acc[0][1] = __builtin_amdgcn_wmma_f32_16x16x4_f32(
          false, a0, false, vb1, (short)0, acc[0][1], false, false);
      acc[1][1] = __builtin_amdgcn_wmma_f32_16x16x4_f32(
          false, a1, false, vb1, (short)0, acc[1][1], false, false);
      acc[0][2] = __builtin_amdgcn_wmma_f32_16x16x4_f32(
          false, a0, false, vb2, (short)0, acc[0][2], false, false);
      acc[1][2] = __builtin_amdgcn_wmma_f32_16x16x4_f32(
          false, a1, false, vb2, (short)0, acc[1][2], false, false);
      acc[0][3] = __builtin_amdgcn_wmma_f32_16x16x4_f32(
          false, a0, false, vb3, (short)0, acc[0][3], false, false);
      acc[1][3] = __builtin_amdgcn_wmma_f32_16x16x4_f32(
          false, a1, false, vb3, (short)0, acc[1][3], false, false);
    }

    // C/D layout: half 0 vgpr v -> row M=v, half 1 -> row M=v+8; col N = lm.
#pragma unroll
    for (int j = 0; j < 4; ++j) {
      int n = (ntb + j) * 16 + lm;
      float es = embsq[n];
#pragma unroll
      for (int r = 0; r < 2; ++r)
#pragma unroll
        for (int v = 0; v < 8; ++v) {
          float d = es - 2.0f * acc[r][j][v];
          if (d < bestv[r][v] || (d == bestv[r][v] && n < besti[r][v])) {
            bestv[r][v] = d;
            besti[r][v] = n;
          }
        }
    }
  }

  // ---- argmin across the 16 lanes of each half (columns) -----------------
#pragma unroll
  for (int r = 0; r < 2; ++r)
#pragma unroll
    for (int v = 0; v < 8; ++v) {
      float bv = bestv[r][v];
      int   bi = besti[r][v];
#pragma unroll
      for (int m = 8; m >= 1; m >>= 1) {  // stays within each 16-lane half
        float ov = __shfl_xor(bv, m, 32);
        int   oi = __shfl_xor(bi, m, 32);
        if (ov < bv || (ov == bv && oi < bi)) { bv = ov; bi = oi; }
      }
      if (lm == 0) {
        red_val[wave][r * 16 + half * 8 + v] = bv;
        red_idx[wave][r * 16 + half * 8 + v] = bi;
      }
    }
  __syncthreads();

  // ---- reduce across the 4 waves; emit indices ---------------------------
  if (tid < ROWS_PER_BLK) {
    float bv = red_val[0][tid];
    int   bi = red_idx[0][tid];
#pragma unroll
    for (int w = 1; w < 4; ++w) {
      float ov = red_val[w][tid];
      int   oi = red_idx[w][tid];
      if (ov < bv || (ov == bv && oi < bi)) { bv = ov; bi = oi; }
    }
    idx_out[n0 + tid]  = bi;
    idxf_out[n0 + tid] = (float)bi;
  }
}

// ---------------------------------------------------------------------------
// Kernel 2: gather z_q = emb[idx] back to [B,C,H,W], accumulate
// loss = (1 + BETA) * mean((z_q - z)^2) = 1.25/N * sum.
// ---------------------------------------------------------------------------
__global__ __launch_bounds__(256) void vq_gather_loss_kernel(
    const float* __restrict__ z, const float* __restrict__ emb,
    const int* __restrict__ idx, float* __restrict__ zq_out,
    float* __restrict__ loss_slot) {
  __shared__ float wave_sums[8];
  const long e = (long)blockIdx.x * 256 + threadIdx.x;  // < N_ELEMS
  // e = b*262144 + c*1024 + hw
  const int hw = (int)(e & 1023);
  const int c  = (int)((e >> 10) & 255);
  const int b  = (int)(e >> 18);
  const int n  = b * 1024 + hw;
  const int k  = idx[n];
  const float zq = emb[(size_t)k * K_DIM + c];
  const float zv = z[e];
  zq_out[e] = zq;  // straight-through forward value == z_q
  float d = zq - zv;
  float p = d * d;
#pragma unroll
  for (int m = 16; m >= 1; m >>= 1) p += __shfl_xor(p, m, 32);
  const int lane = threadIdx.x & 31, wave = threadIdx.x >> 5;
  if (lane == 0) wave_sums[wave] = p;
  __syncthreads();
  if (threadIdx.x == 0) {
    float s = 0.0f;
#pragma unroll
    for (int w = 0; w < 8; ++w) s += wave_sums[w];
    atomicAdd(loss_slot, s * (1.25f / (float)N_ELEMS));
  }
}

// ---------------------------------------------------------------------------
extern "C" void kernel_launch(void* const* d_in, const int* in_sizes, int n_in,
                              void* d_out, int out_size, void* d_ws,
                              size_t ws_size, hipStream_t stream) {
  const float* z   = (const float*)d_in[0];  // [32,256,32,32]
  const float* emb = (const float*)d_in[1];  // [1024,256]

  float* out       = (float*)d_out;
  float* zq_out    = out;                       // 8388608 floats
  float* idxf_out  = out + N_ELEMS;             // 32768 indices (as float)
  float* loss_slot = out + N_ELEMS + N_ROWS;    // 1 scalar

  float* embsq  = (float*)d_ws;                                   // 1024 f32
  int*   idx_ws = (int*)((char*)d_ws + N_EMB * sizeof(float));    // 32768 i32

  vq_embsq_kernel<<<(N_EMB + 255) / 256, 256, 0, stream>>>(emb, embsq,
                                                           loss_slot);
  vq_argmin_kernel<<<N_ROWS / ROWS_PER_BLK, 128, 0, stream>>>(
      z, emb, embsq, idx_ws, idxf_out);
  vq_gather_loss_kernel<<<N_ELEMS / 256, 256, 0, stream>>>(z, emb, idx_ws,
                                                           zq_out, loss_slot);
}